// ReLU_43525198577802
// MI455X (gfx1250) — compile-verified
//
#include <hip/hip_runtime.h>

typedef __attribute__((ext_vector_type(2))) float v2f;
typedef __attribute__((ext_vector_type(4))) float v4f;
typedef __attribute__((ext_vector_type(8))) float v8f;

#define N_ROWS   8192
#define N_COLS   4096
#define OUT_COLS (N_COLS + N_ROWS)   // 12288

// ---------------------------------------------------------------------------
// Kernel 1: zero the eps block (out[:, 4096:12288]) with NT b128 stores.
// ---------------------------------------------------------------------------
__global__ void zero_eps_kernel(v4f* __restrict__ out4) {
    size_t idx = (size_t)blockIdx.x * blockDim.x + threadIdx.x; // 8192*2048 total
    int i  = (int)(idx >> 11);        // row
    int j4 = (int)(idx & 2047);       // float4 within eps block
    v4f z = {0.0f, 0.0f, 0.0f, 0.0f};
    v4f* p = out4 + (size_t)i * (OUT_COLS / 4) + (N_COLS / 4) + j4;
    __builtin_nontemporal_store(z, p);
}

static __device__ inline v2f vabs2(v2f a) {
    a.x = __builtin_fabsf(a.x);
    a.y = __builtin_fabsf(a.y);
    return a;
}

// ---------------------------------------------------------------------------
// Kernel 2: per-16-row stats. 8 waves/block, wave w reduces columns
// [w*512, w*512+512) of 16 rows via V_WMMA_F32_16X16X4_F32 with B = ones.
// A-layout (ISA 7.12.2, 32-bit A 16x4): lane m<16 holds K=0,1; lane m+16
// holds K=2,3 -> per-lane contiguous float2 load at col k + 2*(lane>>4).
// ---------------------------------------------------------------------------
__global__ void rowstats_kernel(const float* __restrict__ layer,
                                const float* __restrict__ lambdas,
                                float* __restrict__ out,
                                float* __restrict__ scale_ws,
                                float* __restrict__ eps_ws) {
    __shared__ float lds[8][16];

    const int rowbase = blockIdx.x * 16;
    const int tid   = threadIdx.x;
    const int wave  = tid >> 5;
    const int lane  = tid & 31;
    const int m     = lane & 15;
    const int khalf = lane >> 4;

    const float* rowptr = layer + (size_t)(rowbase + m) * N_COLS + khalf * 2;
    const int kbase = wave * 512;

    v2f b; b.x = 1.0f; b.y = 1.0f;     // B = ones: layout-independent
    v8f d0 = {}, d1 = {}, d2 = {}, d3 = {};

    for (int k = kbase; k < kbase + 512; k += 16) {
        const float* p = rowptr + k;
        v2f a0 = vabs2(*(const v2f*)(p));
        v2f a1 = vabs2(*(const v2f*)(p + 4));
        v2f a2 = vabs2(*(const v2f*)(p + 8));
        v2f a3 = vabs2(*(const v2f*)(p + 12));
        d0 = __builtin_amdgcn_wmma_f32_16x16x4_f32(false, a0, false, b, (short)0, d0, false, false);
        d1 = __builtin_amdgcn_wmma_f32_16x16x4_f32(false, a1, false, b, (short)0, d1, false, false);
        d2 = __builtin_amdgcn_wmma_f32_16x16x4_f32(false, a2, false, b, (short)0, d2, false, false);
        d3 = __builtin_amdgcn_wmma_f32_16x16x4_f32(false, a3, false, b, (short)0, d3, false, false);
    }
    v8f d = d0 + d1 + d2 + d3;

    // D layout: VGPR v, lanes 0-15 -> row v; lanes 16-31 -> row v+8.
    // Every column equals the row-sum, so lane 0 / lane 16 suffice.
    if (lane == 0) {
        #pragma unroll
        for (int v = 0; v < 8; ++v) lds[wave][v] = d[v];
    }
    if (lane == 16) {
        #pragma unroll
        for (int v = 0; v < 8; ++v) lds[wave][8 + v] = d[v];
    }
    __syncthreads();

    if (tid < 16) {
        const int row = rowbase + tid;
        float s = 0.0f;
        #pragma unroll
        for (int w = 0; w < 8; ++w) s += lds[w][tid];

        const float c   = layer[(size_t)row * N_COLS];
        const float r   = s - __builtin_fabsf(c);   // sum excluded column 0
        const float u   = c + r;
        const float l   = c - r;
        const bool dead     = (u <= 0.0f);
        const bool crossing = (!dead) && (l < 0.0f);
        const float lam = lambdas[row];
        const float thr = u / (u - 1.0f);           // NaN/inf compare -> false, as in JAX
        float ne = (lam >= thr) ? (-l * lam * 0.5f) : (u * (1.0f - lam));
        ne = crossing ? ne : 0.0f;
        const float sc = dead ? 0.0f : (crossing ? lam : 1.0f);

        scale_ws[row] = sc;
        eps_ws[row]   = ne;
        out[(size_t)row * OUT_COLS + N_COLS + row] = ne;  // diagonal (after zero pass)
    }
}

// ---------------------------------------------------------------------------
// Kernel 3: main block: out[i, 0:4096] = layer[i,:]*scale[i] (+eps at col 0).
// layer re-read hits L2 (134 MB < 192 MB); NT stores for the 134 MB output.
// ---------------------------------------------------------------------------
__global__ void scale_main_kernel(const v4f* __restrict__ layer4,
                                  const float* __restrict__ scale_ws,
                                  const float* __restrict__ eps_ws,
                                  float* __restrict__ out) {
    size_t idx = (size_t)blockIdx.x * blockDim.x + threadIdx.x; // 8192*1024 total
    int i  = (int)(idx >> 10);
    int j4 = (int)(idx & 1023);
    v4f v = layer4[(size_t)i * (N_COLS / 4) + j4];
    const float s = scale_ws[i];
    v4f w = v * s;
    if (j4 == 0) w.x += eps_ws[i];
    v4f* p = (v4f*)(out + (size_t)i * OUT_COLS) + j4;
    __builtin_nontemporal_store(w, p);
}

// ---------------------------------------------------------------------------
extern "C" void kernel_launch(void* const* d_in, const int* in_sizes, int n_in,
                              void* d_out, int out_size, void* d_ws, size_t ws_size,
                              hipStream_t stream) {
    const float* layer   = (const float*)d_in[0];
    const float* lambdas = (const float*)d_in[1];
    float* out = (float*)d_out;

    float* scale_ws = (float*)d_ws;          // 8192 floats
    float* eps_ws   = scale_ws + N_ROWS;     // 8192 floats (64 KB total)

    // 1) zero eps block: 8192*2048 float4 stores
    zero_eps_kernel<<<(N_ROWS * (N_ROWS / 4)) / 256, 256, 0, stream>>>((v4f*)out);
    // 2) row stats via WMMA (+ diagonal writes)
    rowstats_kernel<<<N_ROWS / 16, 256, 0, stream>>>(layer, lambdas, out, scale_ws, eps_ws);
    // 3) scaled main block: 8192*1024 float4
    scale_main_kernel<<<(N_ROWS * (N_COLS / 4)) / 256, 256, 0, stream>>>(
        (const v4f*)layer, scale_ws, eps_ws, out);
}